// Decoder_58265526338342
// MI455X (gfx1250) — compile-verified
//
#include <hip/hip_runtime.h>

typedef __attribute__((ext_vector_type(2))) float v2f;
typedef __attribute__((ext_vector_type(4))) float v4f;
typedef __attribute__((ext_vector_type(8))) float v8f;

#define N_NODES 20000
#define E_EDGES 320000
#define NHID    256
#define K_SEL   4096
#define EDGES_PER_BLOCK 32

// ---------------------------------------------------------------------------
// idx = nonzero(labels == 1, size=K_SEL), zero padded. N is small; serial.
// ---------------------------------------------------------------------------
__global__ void build_idx_kernel(const int* __restrict__ labels,
                                 int* __restrict__ idx, int n, int k) {
    if (threadIdx.x == 0 && blockIdx.x == 0) {
        int c = 0;
        for (int i = 0; i < n && c < k; ++i)
            if (labels[i] == 1) idx[c++] = i;
        for (; c < k; ++c) idx[c] = 0;
    }
}

// out[i][j] = b[j]  (seed aggregation buffer with the bias)
__global__ void init_bias_kernel(float* __restrict__ out,
                                 const float* __restrict__ b, int nElems) {
    int i = blockIdx.x * blockDim.x + threadIdx.x;
    if (i < nElems) out[i] = b[i & (NHID - 1)];
}

__global__ void relu_kernel(float* __restrict__ x, int nElems) {
    int i = blockIdx.x * blockDim.x + threadIdx.x;
    if (i < nElems) x[i] = fmaxf(x[i], 0.0f);
}

// ---------------------------------------------------------------------------
// C[nRows x 256] = A[nRows x 256] @ W[256 x 256], fp32 WMMA 16x16x4.
// Block = 8 waves sharing one 64-column strip of W, staged once in LDS
// (pair-K layout: one ds_load_b64 returns the (K,K+1) pair the B-operand
// VGPR layout needs). Each wave computes a 16(M) x 64(N) strip.
// ---------------------------------------------------------------------------
__global__ void __launch_bounds__(256)
gemm256_f32_wmma(const float* __restrict__ A, const float* __restrict__ W,
                 float* __restrict__ C, int nRows) {
    // Bs[k2*128 + n*2 + parity] = W[(2*k2+parity)*256 + nBase + n]
    __shared__ float Bs[128 * 128];   // 64 KB

    const int tid    = threadIdx.x;
    const int lane   = tid & 31;
    const int waveId = tid >> 5;
    const int nBase  = (blockIdx.x & 3) * 64;
    const int mBase  = ((blockIdx.x >> 2) * 8 + waveId) * 16;

    // ---- stage W strip: 4096 float4 loads spread over 256 threads ----
    #pragma unroll
    for (int it = 0; it < 16; ++it) {
        const int j  = tid + it * 256;   // float4 index
        const int k  = j >> 4;           // W row
        const int n4 = (j & 15) * 4;     // strip column
        v4f g = *(const v4f*)(W + (size_t)k * NHID + nBase + n4);
        float* dst = Bs + (k >> 1) * 128 + n4 * 2 + (k & 1);
        dst[0] = g.x; dst[2] = g.y; dst[4] = g.z; dst[6] = g.w;
    }
    __syncthreads();

    if (mBase >= nRows) return;          // wave-uniform; EXEC stays all-1s

    const int half   = lane >> 4;        // 0: K={0,1}, 1: K={2,3}
    const int lane15 = lane & 15;
    const int kOff   = half * 2;
    const float* aPtr = A + (size_t)(mBase + lane15) * NHID + kOff;
    const float* bP0  = Bs + ( 0 + lane15) * 2 + half * 128;
    const float* bP1  = Bs + (16 + lane15) * 2 + half * 128;
    const float* bP2  = Bs + (32 + lane15) * 2 + half * 128;
    const float* bP3  = Bs + (48 + lane15) * 2 + half * 128;

    v8f acc0 = {}, acc1 = {}, acc2 = {}, acc3 = {};

    #pragma unroll 4
    for (int k = 0; k < NHID; k += 4) {
        v2f a  = *(const v2f*)(aPtr + k);
        const int ko = (k >> 1) * 128;   // pair-K row offset
        v2f b0 = *(const v2f*)(bP0 + ko);
        v2f b1 = *(const v2f*)(bP1 + ko);
        v2f b2 = *(const v2f*)(bP2 + ko);
        v2f b3 = *(const v2f*)(bP3 + ko);
        acc0 = __builtin_amdgcn_wmma_f32_16x16x4_f32(false, a, false, b0, (short)0, acc0, false, false);
        acc1 = __builtin_amdgcn_wmma_f32_16x16x4_f32(false, a, false, b1, (short)0, acc1, false, false);
        acc2 = __builtin_amdgcn_wmma_f32_16x16x4_f32(false, a, false, b2, (short)0, acc2, false, false);
        acc3 = __builtin_amdgcn_wmma_f32_16x16x4_f32(false, a, false, b3, (short)0, acc3, false, false);
    }

    const int rowD = mBase + half * 8;
    #pragma unroll
    for (int r = 0; r < 8; ++r) {
        float* c = C + (size_t)(rowD + r) * NHID + nBase + lane15;
        c[0]  = acc0[r];
        c[16] = acc1[r];
        c[32] = acc2[r];
        c[48] = acc3[r];
    }
}

// ---------------------------------------------------------------------------
// Edge-wise aggregation: out[row[e]][f] += support[col[e]][f] * w[e].
// 256 threads = one feature each; support/out live in L2 (20.5 MB each).
// ---------------------------------------------------------------------------
__global__ void spmm_scatter(const float* __restrict__ support,
                             const int* __restrict__ row,
                             const int* __restrict__ col,
                             const float* __restrict__ w,
                             float* __restrict__ out, int nEdges) {
    const int f  = threadIdx.x;
    const int e0 = blockIdx.x * EDGES_PER_BLOCK;
    #pragma unroll 4
    for (int i = 0; i < EDGES_PER_BLOCK; ++i) {
        const int e = e0 + i;
        if (e >= nEdges) return;
        const int r = row[e];
        const int c = col[e];
        const float v = support[(size_t)c * NHID + f] * w[e];
        atomicAdd(out + (size_t)r * NHID + f, v);
    }
}

// out[k][f] = src[idx[k]][f]
__global__ void gather_rows_kernel(const float* __restrict__ src,
                                   const int* __restrict__ idx,
                                   float* __restrict__ out) {
    const int k = blockIdx.x;
    const int f = threadIdx.x;
    out[(size_t)k * NHID + f] = src[(size_t)idx[k] * NHID + f];
}

// ---------------------------------------------------------------------------
// Out[K_SEL x K_SEL] = S @ S^T, S = [K_SEL x 256], fp32 WMMA.
// Block = 8 waves sharing one 64-row strip of S (the S^T columns), staged
// once into LDS in the same pair-K layout as the GEMM.
// ---------------------------------------------------------------------------
__global__ void __launch_bounds__(256)
syrk_f32_wmma(const float* __restrict__ S, float* __restrict__ Out) {
    __shared__ float Bs[128 * 128];   // 64 KB

    const int tid    = threadIdx.x;
    const int lane   = tid & 31;
    const int waveId = tid >> 5;
    const int nBase  = (blockIdx.x & 63) * 64;                 // 64 strips
    const int mBase  = ((blockIdx.x >> 6) * 8 + waveId) * 16;  // 32 supers

    // ---- stage S strip: row n of strip = S[nBase+n][0..255] ----
    #pragma unroll
    for (int it = 0; it < 16; ++it) {
        const int j  = tid + it * 256;   // float4 index
        const int n  = j >> 6;           // strip row (64 float4 per row)
        const int k0 = (j & 63) * 4;
        v4f g = *(const v4f*)(S + (size_t)(nBase + n) * NHID + k0);
        v2f lo = { g.x, g.y };
        v2f hi = { g.z, g.w };
        *(v2f*)(Bs + (k0 >> 1) * 128 + n * 2)       = lo;
        *(v2f*)(Bs + ((k0 >> 1) + 1) * 128 + n * 2) = hi;
    }
    __syncthreads();

    const int half   = lane >> 4;
    const int lane15 = lane & 15;
    const int kOff   = half * 2;
    const float* aPtr = S + (size_t)(mBase + lane15) * NHID + kOff;
    const float* bP0  = Bs + ( 0 + lane15) * 2 + half * 128;
    const float* bP1  = Bs + (16 + lane15) * 2 + half * 128;
    const float* bP2  = Bs + (32 + lane15) * 2 + half * 128;
    const float* bP3  = Bs + (48 + lane15) * 2 + half * 128;

    v8f acc0 = {}, acc1 = {}, acc2 = {}, acc3 = {};

    #pragma unroll 4
    for (int k = 0; k < NHID; k += 4) {
        v2f a  = *(const v2f*)(aPtr + k);
        const int ko = (k >> 1) * 128;
        v2f b0 = *(const v2f*)(bP0 + ko);
        v2f b1 = *(const v2f*)(bP1 + ko);
        v2f b2 = *(const v2f*)(bP2 + ko);
        v2f b3 = *(const v2f*)(bP3 + ko);
        acc0 = __builtin_amdgcn_wmma_f32_16x16x4_f32(false, a, false, b0, (short)0, acc0, false, false);
        acc1 = __builtin_amdgcn_wmma_f32_16x16x4_f32(false, a, false, b1, (short)0, acc1, false, false);
        acc2 = __builtin_amdgcn_wmma_f32_16x16x4_f32(false, a, false, b2, (short)0, acc2, false, false);
        acc3 = __builtin_amdgcn_wmma_f32_16x16x4_f32(false, a, false, b3, (short)0, acc3, false, false);
    }

    const int rowD = mBase + half * 8;
    #pragma unroll
    for (int r = 0; r < 8; ++r) {
        float* c = Out + (size_t)(rowD + r) * K_SEL + nBase + lane15;
        c[0]  = acc0[r];
        c[16] = acc1[r];
        c[32] = acc2[r];
        c[48] = acc3[r];
    }
}

// ---------------------------------------------------------------------------
extern "C" void kernel_launch(void* const* d_in, const int* in_sizes, int n_in,
                              void* d_out, int out_size, void* d_ws, size_t ws_size,
                              hipStream_t stream) {
    const float* X      = (const float*)d_in[0];   // feature_attribute [N,256]
    const int*   erow   = (const int*)  d_in[1];
    const int*   ecol   = (const int*)  d_in[2];
    const float* ew     = (const float*)d_in[3];
    const int*   labels = (const int*)  d_in[4];
    const float* W1     = (const float*)d_in[5];
    const float* b1     = (const float*)d_in[6];
    const float* W2     = (const float*)d_in[7];
    const float* b2     = (const float*)d_in[8];
    const float* Wsd    = (const float*)d_in[9];
    const float* bsd    = (const float*)d_in[10];

    float* out_attr = (float*)d_out;                       // [K_SEL,256]
    float* out_adj  = out_attr + (size_t)K_SEL * NHID;     // [K_SEL,K_SEL]

    float* bufA  = (float*)d_ws;                           // [N,256] GEMM out
    float* bufB  = bufA + (size_t)N_NODES * NHID;          // [N,256] aggregation
    float* s_sel = bufB + (size_t)N_NODES * NHID;          // [K_SEL,256]
    int*   idx   = (int*)(s_sel + (size_t)K_SEL * NHID);   // [K_SEL]

    const int elemsN   = N_NODES * NHID;
    const int ewBlocks = (E_EDGES + EDGES_PER_BLOCK - 1) / EDGES_PER_BLOCK;
    const int mSupers  = (N_NODES / 16 + 7) / 8;           // 157
    const int gemmBlks = mSupers * 4;                      // 628
    const int syrkBlks = (K_SEL / 64) * (K_SEL / 16 / 8);  // 64 * 32 = 2048

    build_idx_kernel<<<1, 32, 0, stream>>>(labels, idx, N_NODES, K_SEL);

    // ---- Attribute decoder layer 1: h = relu(agg(X@W1) + b1) ----
    gemm256_f32_wmma<<<gemmBlks, 256, 0, stream>>>(X, W1, bufA, N_NODES);
    init_bias_kernel<<<elemsN / 256, 256, 0, stream>>>(bufB, b1, elemsN);
    spmm_scatter<<<ewBlocks, NHID, 0, stream>>>(bufA, erow, ecol, ew, bufB, E_EDGES);
    relu_kernel<<<elemsN / 256, 256, 0, stream>>>(bufB, elemsN);

    // ---- Attribute decoder layer 2: agg(h@W2) + b2, select rows ----
    gemm256_f32_wmma<<<gemmBlks, 256, 0, stream>>>(bufB, W2, bufA, N_NODES);
    init_bias_kernel<<<elemsN / 256, 256, 0, stream>>>(bufB, b2, elemsN);
    spmm_scatter<<<ewBlocks, NHID, 0, stream>>>(bufA, erow, ecol, ew, bufB, E_EDGES);
    gather_rows_kernel<<<K_SEL, NHID, 0, stream>>>(bufB, idx, out_attr);

    // ---- Structure decoder: s = (agg(X@Wsd) + bsd)[idx]; out = s@s^T ----
    gemm256_f32_wmma<<<gemmBlks, 256, 0, stream>>>(X, Wsd, bufA, N_NODES);
    init_bias_kernel<<<elemsN / 256, 256, 0, stream>>>(bufB, bsd, elemsN);
    spmm_scatter<<<ewBlocks, NHID, 0, stream>>>(bufA, erow, ecol, ew, bufB, E_EDGES);
    gather_rows_kernel<<<K_SEL, NHID, 0, stream>>>(bufB, idx, s_sel);

    syrk_f32_wmma<<<syrkBlks, 256, 0, stream>>>(s_sel, out_adj);
}